// MacroscopicAggregator_62989990363728
// MI455X (gfx1250) — compile-verified
//
#include <hip/hip_runtime.h>
#include <cstdint>

#define BSZ 256
#define CC  64
#define NN  64
#define PP  128

typedef __attribute__((ext_vector_type(4))) unsigned int u32x4;
typedef __attribute__((ext_vector_type(8))) unsigned int u32x8;

// ---------------------------------------------------------------------------
// Stage 1: one workgroup per (b, n). TDM streams the 64x128 f32 tile
// MIR[b, :, n, :] (row stride 8192 f32) into LDS with +1-dword/row padding.
// Then: residual accumulate + elementwise MLP (in place), metaweight matvec,
// masked softmax, attention matvec. Writes pre-LN x into d_out's MAR region.
// ---------------------------------------------------------------------------
__launch_bounds__(128)
__global__ void macro_agg_stage1(
    const float* __restrict__ MIR,
    const float* __restrict__ covid_mask,
    const float* __restrict__ Wq1, const float* __restrict__ bq1,
    const float* __restrict__ Wq2, const float* __restrict__ bq2,
    const float* __restrict__ metaweights, const float* __restrict__ metabias,
    float* __restrict__ out_x,     // (BS*N, P) pre-LN values (MAR region)
    float* __restrict__ out_attn)  // (BS*N, C)
{
    constexpr int ROW = PP + 1;            // 129 dwords: matches TDM pad
    __shared__ float tile[CC * ROW];       // 33,024 B
    __shared__ float mw_s[PP];
    __shared__ float cm_s[CC];
    __shared__ float lg_s[CC];
    __shared__ float aw_s[CC];

    const int bp = blockIdx.x;             // b' in [0, BS*N)
    const int b  = bp / NN;
    const int n  = bp - b * NN;
    const int t  = threadIdx.x;            // 128 threads = 4 waves

    // ---- wave 0 issues the TDM 2D tile load (EXEC all-ones in this wave)
    if (t < 32) {
        uint64_t gaddr = (uint64_t)(uintptr_t)
            (MIR + ((size_t)b * (CC * NN * PP) + (size_t)n * PP));
        uint32_t lds_off = (uint32_t)(uintptr_t)(&tile[0]); // low 32b = LDS offset

        u32x4 g0;
        g0[0] = 1u;                                          // count=1 (valid)
        g0[1] = lds_off;                                     // lds_addr
        g0[2] = (uint32_t)gaddr;                             // global_addr lo
        g0[3] = ((uint32_t)(gaddr >> 32) & 0x01FFFFFFu)
              | 0x80000000u;                                 // addr hi | type=2

        u32x8 g1;
        g1[0] = (2u << 16)        // data_size = 4 B
              | (1u << 20)        // pad_enable
              | (6u << 22);       // pad_interval = 128 dwords (pad_amount=0 -> 1 dword)
        g1[1] = (uint32_t)PP << 16;   // tensor_dim0 = 128   (bits 79:48, low half)
        g1[2] = (uint32_t)CC << 16;   // tensor_dim1 = 64    (bits 111:80, low half)
        g1[3] = (uint32_t)PP << 16;   // tile_dim0  = 128    (bits 127:112)
        g1[4] = (uint32_t)CC;         // tile_dim1  = 64     (bits 143:128)
        g1[5] = (uint32_t)(NN * PP);  // tensor_dim0_stride = 8192 elems
        g1[6] = 0u;
        g1[7] = 0u;

        asm volatile("tensor_load_to_lds %0, %1" :: "s"(g0), "s"(g1) : "memory");
    }

    // ---- stage small operands while the TDM streams
    mw_s[t] = metaweights[(size_t)((b & (CC - 1)) * NN + n) * PP + t];
    if (t < CC) cm_s[t] = covid_mask[b * CC + t];

    const float wq1a = Wq1[0], wq1b = Wq1[1];
    const float bq1a = bq1[0], bq1b = bq1[1];
    const float wq2a = Wq2[0], wq2b = Wq2[1];
    const float bq2s = bq2[0];
    const float mb   = metabias[(b & (CC - 1)) * NN + n];

    __builtin_amdgcn_s_wait_tensorcnt(0);   // waves w/o TDM pass trivially
    __syncthreads();

    float active = 0.f;
    #pragma unroll
    for (int c = 0; c < CC; ++c) active += cm_s[c];
    const float inv = 1.0f / ((active == 0.f) ? 1.0f : active);

    // ---- fused pass: residual accumulate on raw x, elementwise MLP in place
    float resacc = 0.f;                     // thread t owns column p = t
    #pragma unroll 4
    for (int c = 0; c < CC; ++c) {
        float x  = tile[c * ROW + t];
        resacc   = fmaf(cm_s[c], x, resacc);
        float h1 = fmaxf(fmaf(wq1a, x, bq1a), 0.f);
        float h2 = fmaxf(fmaf(wq1b, x, bq1b), 0.f);
        float m  = fmaxf(fmaf(wq2a, h1, fmaf(wq2b, h2, bq2s)), 0.f);
        tile[c * ROW + t] = m;
    }
    resacc *= inv;
    __syncthreads();

    // ---- attn logits: thread c reads row c (stride 129 -> conflict-free)
    if (t < CC) {
        float s = 0.f;
        #pragma unroll 4
        for (int p = 0; p < PP; ++p) s = fmaf(tile[t * ROW + p], mw_s[p], s);
        s += mb;
        lg_s[t] = (cm_s[t] != 0.f) ? s : -1e30f;
    }
    __syncthreads();

    // ---- masked softmax over 64 entries (broadcast LDS reads, all threads)
    float mx = -__builtin_inff();
    #pragma unroll
    for (int c = 0; c < CC; ++c) mx = fmaxf(mx, lg_s[c]);
    float se = 0.f;
    #pragma unroll
    for (int c = 0; c < CC; ++c) se += __expf(lg_s[c] - mx);
    if (t < CC) {
        float w = (active > 0.f) ? (__expf(lg_s[t] - mx) / se) : 0.f;
        aw_s[t] = w;
        out_attn[(size_t)bp * CC + t] = w;
    }
    __syncthreads();

    // ---- attention-weighted aggregation + residual
    float agg = 0.f;
    #pragma unroll 4
    for (int c = 0; c < CC; ++c) agg = fmaf(tile[c * ROW + t], aw_s[c], agg);
    out_x[(size_t)bp * PP + t] = agg + resacc;
}

// ---------------------------------------------------------------------------
// Stage 2: per-batch LayerNorm over (n,p) = 8192 elems, in place in d_out.
// ---------------------------------------------------------------------------
__launch_bounds__(256)
__global__ void macro_agg_ln(float* __restrict__ xio,
                             const float* __restrict__ ln_w,
                             const float* __restrict__ ln_b)
{
    __shared__ float rs[256];
    __shared__ float rq[256];
    const int b = blockIdx.x, t = threadIdx.x;
    float* x = xio + (size_t)b * (NN * PP);

    float s = 0.f, q = 0.f;
    for (int i = t; i < NN * PP; i += 256) {
        float v = x[i];
        s += v;
        q = fmaf(v, v, q);
    }
    rs[t] = s; rq[t] = q;
    __syncthreads();
    for (int off = 128; off > 0; off >>= 1) {
        if (t < off) { rs[t] += rs[t + off]; rq[t] += rq[t + off]; }
        __syncthreads();
    }
    const float mu  = rs[0] * (1.0f / (NN * PP));
    const float var = rq[0] * (1.0f / (NN * PP)) - mu * mu;
    const float sc  = rsqrtf(var + 1e-5f);

    for (int i = t; i < NN * PP; i += 256) {
        x[i] = (x[i] - mu) * sc * ln_w[i] + ln_b[i];
    }
}

// ---------------------------------------------------------------------------
extern "C" void kernel_launch(void* const* d_in, const int* in_sizes, int n_in,
                              void* d_out, int out_size, void* d_ws, size_t ws_size,
                              hipStream_t stream) {
    const float* MIR  = (const float*)d_in[0];
    // d_in[1]=ECR, d_in[2]=GER, d_in[3]=EPR are unused by the reference
    const float* cmsk = (const float*)d_in[4];
    const float* Wq1  = (const float*)d_in[5];
    const float* bq1  = (const float*)d_in[6];
    const float* Wq2  = (const float*)d_in[7];
    const float* bq2  = (const float*)d_in[8];
    const float* mwt  = (const float*)d_in[9];
    const float* mbi  = (const float*)d_in[10];
    const float* lnw  = (const float*)d_in[11];
    const float* lnb  = (const float*)d_in[12];

    float* out      = (float*)d_out;
    float* out_x    = out;                             // MAR region (in-place LN)
    float* out_attn = out + (size_t)BSZ * NN * PP;     // attn_w region

    macro_agg_stage1<<<BSZ * NN, 128, 0, stream>>>(
        MIR, cmsk, Wq1, bq1, Wq2, bq2, mwt, mbi, out_x, out_attn);
    macro_agg_ln<<<BSZ, 256, 0, stream>>>(out_x, lnw, lnb);
}